// GVAE_61624190763211
// MI455X (gfx1250) — compile-verified
//
#include <hip/hip_runtime.h>
#include <hip/hip_bf16.h>
#include <math.h>

// ---------------- problem constants (from reference) ----------------
#define NG   2048
#define NP   20
#define NN   (NG*NP)        // 40960 nodes
#define EE   163840         // edges
#define FF   30             // input features
#define HH   4              // heads
#define CC   64             // channels per head
#define HC   (HH*CC)        // 256
#define ED   11             // edge feature dim
#define LAT  121
#define DEC  512
#define ATOM_OUT 10
#define EDGE_OUT 950

// ---------------- d_out layout (floats) ----------------
#define OFF_TRIU 0L
#define OFF_NODE 1945600L                 // NG*950
#define OFF_MU   2355200L                 // +N*10
#define OFF_LV   7311360L                 // +N*121

// ---------------- workspace layout (floats) ----------------
// BIG region reused: conv phase Q/K/V; decoder phase A1/A2.
#define WS_Q      0L
#define WS_K      10485760L
#define WS_V      20971520L
#define WS_A1     0L
#define WS_A2     20971520L
#define WS_XR     41943040L               // N*64
#define WS_ALPHA  44564480L               // E*4
#define WS_AMAX   45219840L               // N*4 (ordered-uint)
#define WS_DEN    45383680L               // N*4
#define WS_OUT    45547520L               // N*64
#define WS_H0     48168960L               // N*64
#define WS_H1     50790400L               // N*64
#define WS_Z      53411840L               // N*121
#define WS_U      58368000L               // NG*121
#define WS_BN     58615808L               // 128
#define WS_BETA   58615936L               // N

typedef __attribute__((ext_vector_type(2))) float v2f;
typedef __attribute__((ext_vector_type(8))) float v8f;

// ---------- float <-> order-preserving uint (for atomic max) ----------
__device__ __forceinline__ unsigned f2o(float f) {
    unsigned u = __float_as_uint(f);
    return (u & 0x80000000u) ? ~u : (u | 0x80000000u);
}
__device__ __forceinline__ float o2f(unsigned u) {
    return (u & 0x80000000u) ? __uint_as_float(u ^ 0x80000000u) : __uint_as_float(~u);
}
__device__ __forceinline__ int imin(int a, int b) { return a < b ? a : b; }

// =====================================================================
// Generic fp32 GEMM:  C[M,Nc] = act(A[M,K] @ W[K,Nc] + bias)
// via V_WMMA_F32_16X16X4_F32. Block = 256 thr = 8 waves, tile 128x64.
// Each wave owns a 32x32 tile: 4 accumulators, A frags reused across N,
// B frags reused across M. Main K loop is UNGUARDED: OOB rows/cols load
// from clamped addresses and accumulate into lanes the (guarded) store
// phase never writes; only the K%4 epilogue needs zero-masking, since
// the K dimension reduces into every output. All addressing is
// strength-reduced to incrementing pointers (no per-iter multiplies).
// =====================================================================
__global__ __launch_bounds__(256)
void gemm_f32_wmma(const float* __restrict__ A, const float* __restrict__ W,
                   const float* __restrict__ bias, float* __restrict__ C,
                   int M, int K, int Nc, int act) {
    const int lane = threadIdx.x & 31;
    const int wave = threadIdx.x >> 5;     // 0..7
    const int wm = wave >> 1;              // 0..3  (M subtiles of 32)
    const int wn = wave & 1;               // 0..1  (N subtiles of 32)
    const int m0 = blockIdx.y * 128 + wm * 32;
    const int n0 = blockIdx.x * 64 + wn * 32;
    const int hl = lane >> 4;              // lane half
    const int ll = lane & 15;

    // clamped A row bases (two M subtiles) and W column indices (two N subtiles)
    const int ar0 = m0 + ll;
    const int ar1 = m0 + 16 + ll;
    const long ab0 = (long)imin(ar0, M - 1) * K;
    const long ab1 = (long)imin(ar1, M - 1) * K;
    const int c0 = n0 + ll;
    const int c1 = n0 + 16 + ll;
    const int c0c = imin(c0, Nc - 1);
    const int c1c = imin(c1, Nc - 1);

    v8f acc00 = {}, acc01 = {}, acc10 = {}, acc11 = {};
    const int Kmain = K & ~3;
    const long ldw = (long)Nc;

    // incrementing pointers: A frag rows hold K {k+2*hl, k+1+2*hl};
    // B frag regs hold rows {k+hl, k+2+hl} at fixed columns.
    const float* pA0  = A + ab0 + 2 * hl;
    const float* pA1  = A + ab1 + 2 * hl;
    const float* pB0l = W + (long)hl * ldw + c0c;       // row k+hl,   col c0
    const float* pB0h = pB0l + 2 * ldw;                 // row k+2+hl, col c0
    const float* pB1l = W + (long)hl * ldw + c1c;       // row k+hl,   col c1
    const float* pB1h = pB1l + 2 * ldw;                 // row k+2+hl, col c1
    const long bstep = 4 * ldw;

#pragma unroll 2
    for (int k = 0; k < Kmain; k += 4) {
        v2f a0, a1, b0, b1;
        a0.x = pA0[0];  a0.y = pA0[1];
        a1.x = pA1[0];  a1.y = pA1[1];
        b0.x = pB0l[0]; b0.y = pB0h[0];
        b1.x = pB1l[0]; b1.y = pB1h[0];
        acc00 = __builtin_amdgcn_wmma_f32_16x16x4_f32(false, a0, false, b0,
                                                      (short)0, acc00, false, false);
        acc01 = __builtin_amdgcn_wmma_f32_16x16x4_f32(false, a0, false, b1,
                                                      (short)0, acc01, false, false);
        acc10 = __builtin_amdgcn_wmma_f32_16x16x4_f32(false, a1, false, b0,
                                                      (short)0, acc10, false, false);
        acc11 = __builtin_amdgcn_wmma_f32_16x16x4_f32(false, a1, false, b1,
                                                      (short)0, acc11, false, false);
        pA0 += 4; pA1 += 4;
        pB0l += bstep; pB0h += bstep; pB1l += bstep; pB1h += bstep;
    }

    if (Kmain < K) {   // epilogue: mask out k >= K contributions
        const int k = Kmain;
        const int ka = k + 2 * hl;
        const float gx = (ka     < K) ? 1.f : 0.f;
        const float gy = (ka + 1 < K) ? 1.f : 0.f;
        const int kax = imin(ka, K - 1);
        const int kay = imin(ka + 1, K - 1);
        v2f a0, a1;
        a0.x = A[ab0 + kax] * gx;  a0.y = A[ab0 + kay] * gy;
        a1.x = A[ab1 + kax] * gx;  a1.y = A[ab1 + kay] * gy;
        const int r0 = k + hl, r1 = k + 2 + hl;
        const float h0 = (r0 < K) ? 1.f : 0.f;
        const float h1 = (r1 < K) ? 1.f : 0.f;
        const long kb0 = (long)imin(r0, K - 1) * ldw;
        const long kb1 = (long)imin(r1, K - 1) * ldw;
        v2f b0, b1;
        b0.x = W[kb0 + c0c] * h0;  b0.y = W[kb1 + c0c] * h1;
        b1.x = W[kb0 + c1c] * h0;  b1.y = W[kb1 + c1c] * h1;
        acc00 = __builtin_amdgcn_wmma_f32_16x16x4_f32(false, a0, false, b0,
                                                      (short)0, acc00, false, false);
        acc01 = __builtin_amdgcn_wmma_f32_16x16x4_f32(false, a0, false, b1,
                                                      (short)0, acc01, false, false);
        acc10 = __builtin_amdgcn_wmma_f32_16x16x4_f32(false, a1, false, b0,
                                                      (short)0, acc10, false, false);
        acc11 = __builtin_amdgcn_wmma_f32_16x16x4_f32(false, a1, false, b1,
                                                      (short)0, acc11, false, false);
    }

    // C/D layout: VGPR r -> M = tile_m + r (+8 for upper lane half), N = tile_n + ll
    const float bv0 = (c0 < Nc) ? bias[c0] : 0.f;
    const float bv1 = (c1 < Nc) ? bias[c1] : 0.f;
    const int rb = hl ? 8 : 0;
#pragma unroll
    for (int r = 0; r < 8; ++r) {
        const int row0 = m0 + rb + r;
        const int row1 = m0 + 16 + rb + r;
        if (row0 < M) {
            if (c0 < Nc) {
                float v = acc00[r] + bv0;
                if (act) v = fmaxf(v, 0.f);
                C[(long)row0 * Nc + c0] = v;
            }
            if (c1 < Nc) {
                float v = acc01[r] + bv1;
                if (act) v = fmaxf(v, 0.f);
                C[(long)row0 * Nc + c1] = v;
            }
        }
        if (row1 < M) {
            if (c0 < Nc) {
                float v = acc10[r] + bv0;
                if (act) v = fmaxf(v, 0.f);
                C[(long)row1 * Nc + c0] = v;
            }
            if (c1 < Nc) {
                float v = acc11[r] + bv1;
                if (act) v = fmaxf(v, 0.f);
                C[(long)row1 * Nc + c1] = v;
            }
        }
    }
}

// ---------------- small elementwise / scatter kernels ----------------
__global__ void fill_kernel(float* p, float v, long n) {
    long i = (long)blockIdx.x * blockDim.x + threadIdx.x;
    if (i < n) p[i] = v;
}

// attention logits: alpha[e,h] = <q[dst,h,:], k[src,h,:] + e_mlp(e,h,:)> / 8
__global__ void alpha_kernel(const float* __restrict__ ea, const float* __restrict__ We,
                             const float* __restrict__ be, const float* __restrict__ Q,
                             const float* __restrict__ K, const int* __restrict__ src,
                             const int* __restrict__ dst, float* __restrict__ Alpha,
                             unsigned* __restrict__ Amax) {
    long idx = (long)blockIdx.x * blockDim.x + threadIdx.x;
    if (idx >= (long)EE * HH) return;
    int e = (int)(idx >> 2), h = (int)(idx & 3);
    int s = src[e], d = dst[e];
    const float* eav = ea + (long)e * ED;
    float ec0[ED];
#pragma unroll
    for (int j = 0; j < ED; ++j) ec0[j] = eav[j];
    float sum = 0.f;
    for (int c = 0; c < CC; ++c) {
        int hc = h * CC + c;
        float ec = be[hc];
#pragma unroll
        for (int j = 0; j < ED; ++j) ec += ec0[j] * We[j * HC + hc];
        sum += Q[(long)d * HC + hc] * (K[(long)s * HC + hc] + ec);
    }
    float a = sum * 0.125f;   // 1/sqrt(64)
    Alpha[idx] = a;
    atomicMax(&Amax[(long)d * HH + h], f2o(a));
}

// ex = exp(alpha - amax[dst]); den[dst] += ex
__global__ void exden_kernel(float* __restrict__ Alpha, const unsigned* __restrict__ Amax,
                             float* __restrict__ Den, const int* __restrict__ dst) {
    long idx = (long)blockIdx.x * blockDim.x + threadIdx.x;
    if (idx >= (long)EE * HH) return;
    int e = (int)(idx >> 2), h = (int)(idx & 3);
    int d = dst[e];
    float ex = expf(Alpha[idx] - o2f(Amax[(long)d * HH + h]));
    Alpha[idx] = ex;
    atomicAdd(&Den[(long)d * HH + h], ex);
}

// out[dst,c] += (1/H) * sum_h a[e,h] * (v[src,h,c] + e_mlp(e,h,c))
__global__ void msg_kernel(const float* __restrict__ ea, const float* __restrict__ We,
                           const float* __restrict__ be, const float* __restrict__ V,
                           const float* __restrict__ Ex, const float* __restrict__ Den,
                           const int* __restrict__ src, const int* __restrict__ dst,
                           float* __restrict__ Out) {
    long idx = (long)blockIdx.x * blockDim.x + threadIdx.x;
    if (idx >= (long)EE * CC) return;
    int e = (int)(idx >> 6), c = (int)(idx & 63);
    int s = src[e], d = dst[e];
    const float* eav = ea + (long)e * ED;
    float ec0[ED];
#pragma unroll
    for (int j = 0; j < ED; ++j) ec0[j] = eav[j];
    float acc = 0.f;
#pragma unroll
    for (int h = 0; h < HH; ++h) {
        int hc = h * CC + c;
        float ec = be[hc];
#pragma unroll
        for (int j = 0; j < ED; ++j) ec += ec0[j] * We[j * HC + hc];
        float coef = Ex[(long)e * HH + h] / Den[(long)d * HH + h];
        acc += coef * (V[(long)s * HC + hc] + ec);
    }
    atomicAdd(&Out[(long)d * CC + c], acc * 0.25f);
}

// beta = sigmoid([out, xr, out-xr] @ Wb)
__global__ void beta_kernel(const float* __restrict__ Out, const float* __restrict__ Xr,
                            const float* __restrict__ Wb, float* __restrict__ Beta) {
    int n = blockIdx.x * blockDim.x + threadIdx.x;
    if (n >= NN) return;
    const float* o = Out + (long)n * CC;
    const float* x = Xr + (long)n * CC;
    float s = 0.f;
    for (int c = 0; c < CC; ++c) {
        float ov = o[c], xv = x[c];
        s += ov * Wb[c] + xv * Wb[CC + c] + (ov - xv) * Wb[2 * CC + c];
    }
    Beta[n] = 1.f / (1.f + expf(-s));
}

// h = relu(beta*xr + (1-beta)*out)
__global__ void combine_kernel(const float* __restrict__ Out, const float* __restrict__ Xr,
                               const float* __restrict__ Beta, float* __restrict__ H) {
    long i = (long)blockIdx.x * blockDim.x + threadIdx.x;
    if (i >= (long)NN * CC) return;
    float b = Beta[i >> 6];
    H[i] = fmaxf(b * Xr[i] + (1.f - b) * Out[i], 0.f);
}

// batchnorm pass 1: per-channel mean / biased var (one block per channel)
__global__ __launch_bounds__(256)
void bn_stats_kernel(const float* __restrict__ H, float* __restrict__ stats) {
    int c = blockIdx.x, tid = threadIdx.x;
    float s = 0.f, sq = 0.f;
    for (int n = tid; n < NN; n += 256) {
        float v = H[(long)n * CC + c];
        s += v; sq += v * v;
    }
    __shared__ float rs[256], rq[256];
    rs[tid] = s; rq[tid] = sq; __syncthreads();
    for (int st = 128; st > 0; st >>= 1) {
        if (tid < st) { rs[tid] += rs[tid + st]; rq[tid] += rq[tid + st]; }
        __syncthreads();
    }
    if (tid == 0) {
        float m = rs[0] / (float)NN;
        stats[c] = m;
        stats[CC + c] = rq[0] / (float)NN - m * m;
    }
}

__global__ void bn_apply_kernel(float* __restrict__ H, const float* __restrict__ stats,
                                const float* __restrict__ g, const float* __restrict__ b) {
    long i = (long)blockIdx.x * blockDim.x + threadIdx.x;
    if (i >= (long)NN * CC) return;
    int c = (int)(i & 63);
    H[i] = (H[i] - stats[c]) * rsqrtf(stats[CC + c] + 1e-5f) * g[c] + b[c];
}

// z = eps * exp(logvar) + mu
__global__ void z_kernel(const float* __restrict__ eps, const float* __restrict__ lv,
                         const float* __restrict__ mu, float* __restrict__ Z) {
    long i = (long)blockIdx.x * blockDim.x + threadIdx.x;
    if (i >= (long)NN * LAT) return;
    Z[i] = eps[i] * expf(lv[i]) + mu[i];
}

// per-graph: u_g = sum_m w_m * z[g,m,:], w = column means of normalized adj
__global__ __launch_bounds__(128)
void graph_u_kernel(const float* __restrict__ Z, float* __restrict__ U) {
    __shared__ float zb[NP * LAT];
    __shared__ float sA[NP * NP];
    __shared__ float sdis[NP];
    __shared__ float sw[NP];
    __shared__ float red[128];
    const int g = blockIdx.x, tid = threadIdx.x;
    const float* zg = Z + (long)g * NP * LAT;
    for (int i = tid; i < NP * LAT; i += 128) zb[i] = zg[i];
    __syncthreads();
    float p = 0.f;
    for (int i = tid; i < NP * LAT; i += 128) p += zb[i] * zb[i];
    red[tid] = p; __syncthreads();
    for (int s = 64; s > 0; s >>= 1) {
        if (tid < s) red[tid] += red[tid + s];
        __syncthreads();
    }
    float fro = red[0];
    for (int idx = tid; idx < NP * NP; idx += 128) {
        int n = idx / NP, m = idx % NP;
        float v;
        if (n == m) v = 1.f;
        else {
            float d = 0.f;
            for (int k = 0; k < LAT; ++k) d += zb[n * LAT + k] * zb[m * LAT + k];
            v = d / fro + 1.f;
        }
        sA[idx] = v;
    }
    __syncthreads();
    if (tid < NP) {
        float d = 0.f;
        for (int m = 0; m < NP; ++m) d += sA[tid * NP + m];
        sdis[tid] = rsqrtf(d);
    }
    __syncthreads();
    if (tid < NP) {
        float wv = 0.f;
        for (int n = 0; n < NP; ++n) wv += sdis[n] * sA[n * NP + tid];
        sw[tid] = wv * sdis[tid] * (1.f / (float)NP);
    }
    __syncthreads();
    for (int k = tid; k < LAT; k += 128) {
        float u = 0.f;
        for (int m = 0; m < NP; ++m) u += sw[m] * zb[m * LAT + k];
        U[(long)g * LAT + k] = u;
    }
}

// ---------------- host side ----------------
static inline void launch_gemm(const float* A, const float* W, const float* b, float* C,
                               int M, int K, int Nc, int act, hipStream_t st) {
    dim3 grid((Nc + 63) / 64, (M + 127) / 128);
    gemm_f32_wmma<<<grid, 256, 0, st>>>(A, W, b, C, M, K, Nc, act);
}

struct ConvP { const float *qW,*qb,*kW,*kb,*vW,*vb,*eW,*eb,*sW,*sb,*bW; };

extern "C" void kernel_launch(void* const* d_in, const int* in_sizes, int n_in,
                              void* d_out, int out_size, void* d_ws, size_t ws_size,
                              hipStream_t stream) {
    (void)in_sizes; (void)n_in; (void)out_size; (void)ws_size;

    // inputs in setup_inputs() dict insertion order, params flattened in
    // insertion order: conv1..4 (q.W,q.b,k.W,k.b,v.W,v.b,e.W,e.b,skip.W,skip.b,beta.W),
    // bn1..3 (g,b), mu, logvar, lin1, lin2, atom, gcn (W,b each)
    const float* x   = (const float*)d_in[0];
    const float* ea  = (const float*)d_in[1];
    const int*   ei  = (const int*)  d_in[2];
    const float* eps = (const float*)d_in[4];
    const int* src = ei;
    const int* dst = ei + EE;

    int p = 5;
    ConvP conv[4];
    for (int l = 0; l < 4; ++l) {
        conv[l].qW = (const float*)d_in[p++]; conv[l].qb = (const float*)d_in[p++];
        conv[l].kW = (const float*)d_in[p++]; conv[l].kb = (const float*)d_in[p++];
        conv[l].vW = (const float*)d_in[p++]; conv[l].vb = (const float*)d_in[p++];
        conv[l].eW = (const float*)d_in[p++]; conv[l].eb = (const float*)d_in[p++];
        conv[l].sW = (const float*)d_in[p++]; conv[l].sb = (const float*)d_in[p++];
        conv[l].bW = (const float*)d_in[p++];
    }
    const float* bng[3]; const float* bnb[3];
    for (int l = 0; l < 3; ++l) { bng[l] = (const float*)d_in[p++]; bnb[l] = (const float*)d_in[p++]; }
    const float* muW  = (const float*)d_in[p++]; const float* mub  = (const float*)d_in[p++];
    const float* lvW  = (const float*)d_in[p++]; const float* lvb  = (const float*)d_in[p++];
    const float* l1W  = (const float*)d_in[p++]; const float* l1b  = (const float*)d_in[p++];
    const float* l2W  = (const float*)d_in[p++]; const float* l2b  = (const float*)d_in[p++];
    const float* atW  = (const float*)d_in[p++]; const float* atb  = (const float*)d_in[p++];
    const float* gcW  = (const float*)d_in[p++]; const float* gcb  = (const float*)d_in[p++];

    float* w = (float*)d_ws;
    float* out = (float*)d_out;
    float* Qb = w + WS_Q;  float* Kb = w + WS_K;  float* Vb = w + WS_V;
    float* Xr = w + WS_XR; float* Alpha = w + WS_ALPHA;
    float* Amax = w + WS_AMAX; float* Den = w + WS_DEN; float* Outb = w + WS_OUT;
    float* Zb = w + WS_Z;  float* Ub = w + WS_U; float* BNs = w + WS_BN;
    float* Beta = w + WS_BETA;
    float* A1 = w + WS_A1; float* A2 = w + WS_A2;

    const long EH  = (long)EE * HH;
    const long EC  = (long)EE * CC;
    const long NC_ = (long)NN * CC;
    const long NL  = (long)NN * LAT;

    // ---------------- 4 TransformerConv layers ----------------
    const float* cur = x;
    int fi = FF;
    for (int l = 0; l < 4; ++l) {
        launch_gemm(cur, conv[l].qW, conv[l].qb, Qb, NN, fi, HC, 0, stream);
        launch_gemm(cur, conv[l].kW, conv[l].kb, Kb, NN, fi, HC, 0, stream);
        launch_gemm(cur, conv[l].vW, conv[l].vb, Vb, NN, fi, HC, 0, stream);
        launch_gemm(cur, conv[l].sW, conv[l].sb, Xr, NN, fi, CC, 0, stream);
        // zero Amax (ordered-uint 0 == minimum), Den, Out (contiguous region)
        {
            long n = (long)NN * HH * 2 + NC_;
            fill_kernel<<<(unsigned)((n + 255) / 256), 256, 0, stream>>>(Amax, 0.f, n);
        }
        alpha_kernel<<<(unsigned)((EH + 255) / 256), 256, 0, stream>>>(
            ea, conv[l].eW, conv[l].eb, Qb, Kb, src, dst, Alpha, (unsigned*)Amax);
        exden_kernel<<<(unsigned)((EH + 255) / 256), 256, 0, stream>>>(
            Alpha, (const unsigned*)Amax, Den, dst);
        msg_kernel<<<(unsigned)((EC + 255) / 256), 256, 0, stream>>>(
            ea, conv[l].eW, conv[l].eb, Vb, Alpha, Den, src, dst, Outb);
        beta_kernel<<<(NN + 255) / 256, 256, 0, stream>>>(Outb, Xr, conv[l].bW, Beta);
        float* Hout = (l & 1) ? (w + WS_H1) : (w + WS_H0);
        combine_kernel<<<(unsigned)((NC_ + 255) / 256), 256, 0, stream>>>(Outb, Xr, Beta, Hout);
        if (l < 3) {
            bn_stats_kernel<<<CC, 256, 0, stream>>>(Hout, BNs);
            bn_apply_kernel<<<(unsigned)((NC_ + 255) / 256), 256, 0, stream>>>(Hout, BNs, bng[l], bnb[l]);
        }
        cur = Hout;
        fi = CC;
    }

    // ---------------- VAE head ----------------
    launch_gemm(cur, muW, mub, out + OFF_MU, NN, CC, LAT, 0, stream);
    launch_gemm(cur, lvW, lvb, out + OFF_LV, NN, CC, LAT, 0, stream);
    z_kernel<<<(unsigned)((NL + 255) / 256), 256, 0, stream>>>(eps, out + OFF_LV, out + OFF_MU, Zb);

    // ---------------- atom decoder ----------------
    launch_gemm(Zb, l1W, l1b, A1, NN, LAT, DEC, 1, stream);
    launch_gemm(A1, l2W, l2b, A2, NN, DEC, DEC, 1, stream);
    launch_gemm(A2, atW, atb, out + OFF_NODE, NN, DEC, ATOM_OUT, 0, stream);

    // ---------------- edge decoder (mean folded into GCN input) ----------------
    graph_u_kernel<<<NG, 128, 0, stream>>>(Zb, Ub);
    launch_gemm(Ub, gcW, gcb, out + OFF_TRIU, NG, LAT, EDGE_OUT, 0, stream);
}